// PokerAttention_27066883899885
// MI455X (gfx1250) — compile-verified
//
#include <hip/hip_runtime.h>
#include <hip/hip_bf16.h>
#include <stdint.h>
#include <stddef.h>

// ---------------------------------------------------------------------------
// PokerAttention forward for MI455X (gfx1250), fp32 end-to-end via
// V_WMMA_F32_16X16X4_F32. Bandwidth-bound on the 256MB attn output
// (~11us floor @ 23.3 TB/s); compute (~34 GFLOP) hides under that, so fp32
// matrix cores preserve reference accuracy at zero perf cost.
// This revision: b128 NT stores + float4 softmax sweeps on the dominant
// stream, async global->LDS staging (ASYNCcnt) for the bias tables, and
// global_prefetch on the K/V streams.
// ---------------------------------------------------------------------------

typedef float v2f __attribute__((ext_vector_type(2)));
typedef float v4f __attribute__((ext_vector_type(4)));
typedef float v8f __attribute__((ext_vector_type(8)));
typedef short v4s __attribute__((ext_vector_type(4)));

#define S_LEN   2048
#define DM      1024
#define N_HEADS 16
#define DK      64
#define SPAD    2052   // LDS row stride (floats): +4 banks/row -> conflict-free

// ---------------------------------------------------------------------------
// C[M,N] = A[M,K] * B[N,K]^T + bias[N]   (all row-major fp32)
// One 16x16 C tile per wave, 8 waves per 256-thread block.
// WMMA f32 16x16x4 fragment addressing (ISA 7.12.2, 32-bit A 16x4):
//   lane l (0..15): VGPR0=A[l][k], VGPR1=A[l][k+1]; lanes 16..31: k+2,k+3
//   -> per-lane float2 load at  row=(base+(l&15)), col=k+2*(l>>4)
// ---------------------------------------------------------------------------
__global__ __launch_bounds__(256) void gemm_bias_kernel(
    const float* __restrict__ A, const float* __restrict__ B,
    const float* __restrict__ bias, float* __restrict__ C,
    int M, int N, int K)
{
    const int lane = threadIdx.x & 31;
    const int wave = threadIdx.x >> 5;
    const int l15  = lane & 15;
    const int half = lane >> 4;

    const int tiles_n = N >> 4;
    const int tile = blockIdx.x * 8 + wave;
    const int mt = tile / tiles_n;
    const int nt = tile - mt * tiles_n;
    if (mt >= (M >> 4)) return;            // wave-uniform: EXEC stays all-1s

    const float* arow = A + (size_t)(mt * 16 + l15) * K + 2 * half;
    const float* brow = B + (size_t)(nt * 16 + l15) * K + 2 * half;

    // C layout: VGPR i, lanes 0-15 -> M=i, N=lane; lanes 16-31 -> M=8+i
    v8f acc;
    {
        const float bv = bias[nt * 16 + l15];
#pragma unroll
        for (int i = 0; i < 8; ++i) acc[i] = bv;
    }

    for (int k = 0; k < K; k += 4) {
        v2f a = *(const v2f*)(arow + k);
        v2f b = *(const v2f*)(brow + k);
        acc = __builtin_amdgcn_wmma_f32_16x16x4_f32(false, a, false, b,
                                                    (short)0, acc, false, false);
    }

    float* crow = C + (size_t)(mt * 16 + half * 8) * N + nt * 16 + l15;
#pragma unroll
    for (int i = 0; i < 8; ++i) crow[(size_t)i * N] = acc[i];
}

// ---------------------------------------------------------------------------
// Attention core: one workgroup per (head, 16-row query block).
// LDS: 16 x SPAD score/prob block (padded) + per-key bias tables. The raw
// card/pos/mask arrays are async-DMA'd into the (not-yet-used) score region,
// converted to tables, then the score block overwrites the staging area.
// ---------------------------------------------------------------------------
__global__ __launch_bounds__(256) void attn_kernel(
    const float* __restrict__ Q, const float* __restrict__ Km,
    const float* __restrict__ V,
    const int* __restrict__ card, const int* __restrict__ pos,
    const uint8_t* __restrict__ mask,      // numpy bool -> 1 byte/elem
    float* __restrict__ attn, float* __restrict__ ctx)
{
    extern __shared__ char smem_raw[];
    float* sP    = (float*)smem_raw;                         // [16][SPAD]
    short* sRank = (short*)(smem_raw + (size_t)16 * SPAD * 4);
    short* sSuit = sRank + S_LEN;
    short* sPos  = sSuit + S_LEN;
    float* sMask = (float*)(sPos + S_LEN);                   // mask -> -1e9/0

    const int lane = threadIdx.x & 31;
    const int wave = threadIdx.x >> 5;
    const int l15  = lane & 15;
    const int half = lane >> 4;
    const int qbase = blockIdx.x * 16;
    const int head  = blockIdx.y;
    const int hq    = head * DK;

    // ---- async-stage raw card/pos/mask into LDS offset 0 (score region) ----
    // VDST operand = LDS byte offset within our allocation; VADDR = 64b global.
    {
        const int t = threadIdx.x;
#pragma unroll
        for (int off = t * 16; off < 8192; off += 256 * 16) {
            uint32_t lds_off = (uint32_t)off;                     // card @ 0
            uint64_t ga = (uint64_t)(uintptr_t)((const char*)card + off);
            asm volatile("global_load_async_to_lds_b128 %0, %1, off"
                         :: "v"(lds_off), "v"(ga) : "memory");
        }
#pragma unroll
        for (int off = t * 16; off < 8192; off += 256 * 16) {
            uint32_t lds_off = (uint32_t)(8192 + off);            // pos @ 8K
            uint64_t ga = (uint64_t)(uintptr_t)((const char*)pos + off);
            asm volatile("global_load_async_to_lds_b128 %0, %1, off"
                         :: "v"(lds_off), "v"(ga) : "memory");
        }
        {
            uint32_t lds_off = (uint32_t)(16384 + t * 8);         // mask @ 16K
            uint64_t ga = (uint64_t)(uintptr_t)((const char*)mask + t * 8);
            asm volatile("global_load_async_to_lds_b64 %0, %1, off"
                         :: "v"(lds_off), "v"(ga) : "memory");
        }
    }

    // ---- overlap: preload this wave's Q fragments while the DMA flies ----
    const float* qrowp = Q + (size_t)(qbase + l15) * DM + hq + 2 * half;
    v2f qf[16];
#pragma unroll
    for (int j = 0; j < 16; ++j) qf[j] = *(const v2f*)(qrowp + 4 * j);

    asm volatile("s_wait_asynccnt 0x0" ::: "memory");
    __syncthreads();

    // ---- convert raw staging -> bias tables (tables live past sP) ----
    {
        const int*     rawCard = (const int*)smem_raw;
        const int*     rawPos  = (const int*)(smem_raw + 8192);
        const uint8_t* rawMask = (const uint8_t*)(smem_raw + 16384);
        for (int k = threadIdx.x; k < S_LEN; k += 256) {
            const int c = rawCard[k];
            sRank[k] = (short)(c % 13);
            sSuit[k] = (short)(c / 13);
            sPos[k]  = (short)rawPos[k];
            sMask[k] = rawMask[k] ? -1.0e9f : 0.0f;
        }
    }
    __syncthreads();

    // ---- phase 1: raw scores (Q K^T)/sqrt(dk) -> LDS ----
    for (int t = wave; t < S_LEN / 16; t += 8) {
        const int kb = t * 16;
        const float* krowp = Km + (size_t)(kb + l15) * DM + hq + 2 * half;
        // prefetch this wave's next tile (speculative; OOB safely dropped)
        __builtin_prefetch((const void*)(krowp + (size_t)128 * DM), 0, 0);
        v8f acc = {};
#pragma unroll
        for (int j = 0; j < 16; ++j) {
            v2f b = *(const v2f*)(krowp + 4 * j);
            acc = __builtin_amdgcn_wmma_f32_16x16x4_f32(false, qf[j], false, b,
                                                        (short)0, acc, false, false);
        }
#pragma unroll
        for (int i = 0; i < 8; ++i)
            sP[(half * 8 + i) * SPAD + kb + l15] = acc[i] * 0.125f; // 1/sqrt(64)
    }
    __syncthreads();

    // ---- phase 2: bias + mask + softmax; b128 NT-stream attn to HBM ----
    for (int r = wave * 2; r < wave * 2 + 2; ++r) {
        const int qrow = qbase + r;
        const int cq = card[qrow];
        const int rankq = cq % 13, suitq = cq / 13;
        const int pq = pos[qrow];
        float* row = sP + r * SPAD;

        float m = -3.402823466e38f;
        for (int it = 0; it < S_LEN / 128; ++it) {
            const int k = it * 128 + lane * 4;
            v4f s  = *(v4f*)(row + k);
            v4s rk = *(const v4s*)(sRank + k);
            v4s st = *(const v4s*)(sSuit + k);
            v4s ps = *(const v4s*)(sPos + k);
            v4f mk = *(const v4f*)(sMask + k);
#pragma unroll
            for (int j = 0; j < 4; ++j) {
                int rd = rankq - (int)rk[j]; rd = rd < 0 ? -rd : rd;
                float b = (rd == 0) ? 0.2f : ((rd == 1) ? 0.1f : 0.0f);
                if (suitq == (int)st[j]) b += 0.3f;
                int pd = pq - (int)ps[j]; pd = pd < 0 ? -pd : pd;
                if (pd <= 2) b += 0.1f;
                s[j] += b + mk[j];
                m = fmaxf(m, s[j]);
            }
            *(v4f*)(row + k) = s;
        }
#pragma unroll
        for (int off = 16; off > 0; off >>= 1) m = fmaxf(m, __shfl_xor(m, off, 32));

        float sum = 0.0f;
        for (int it = 0; it < S_LEN / 128; ++it) {
            const int k = it * 128 + lane * 4;
            v4f s = *(v4f*)(row + k);
#pragma unroll
            for (int j = 0; j < 4; ++j) { s[j] = __expf(s[j] - m); sum += s[j]; }
            *(v4f*)(row + k) = s;
        }
#pragma unroll
        for (int off = 16; off > 0; off >>= 1) sum += __shfl_xor(sum, off, 32);
        const float inv = 1.0f / sum;

        float* arow_g = attn + ((size_t)(head * S_LEN + qrow)) * S_LEN;
        for (int it = 0; it < S_LEN / 128; ++it) {
            const int k = it * 128 + lane * 4;
            v4f p = *(v4f*)(row + k) * inv;
            *(v4f*)(row + k) = p;
            __builtin_nontemporal_store(p, (v4f*)(arow_g + k)); // b128 NT store
        }
    }
    __syncthreads();

    // ---- phase 3: ctx[16, dk] = P x V ; waves 0-3 take 16-col slabs ----
    if (wave < 4) {
        const int dbase = hq + wave * 16;
        const float* vcol = V + (size_t)(2 * half) * DM + dbase + l15;
        const float* prow = sP + (size_t)l15 * SPAD + 2 * half;
        v8f acc = {};
        for (int kb0 = 0; kb0 < S_LEN; kb0 += 64) {
            // prefetch next 64-row V slab (speculative; OOB dropped)
            __builtin_prefetch((const void*)(vcol + (size_t)(kb0 + 64) * DM), 0, 0);
#pragma unroll
            for (int kb = kb0; kb < kb0 + 64; kb += 4) {
                v2f a = *(const v2f*)(prow + kb);        // LDS, conflict-free
                const float* vp = vcol + (size_t)kb * DM;
                v2f b; b.x = vp[0]; b.y = vp[DM];
                acc = __builtin_amdgcn_wmma_f32_16x16x4_f32(false, a, false, b,
                                                            (short)0, acc, false, false);
            }
        }
        float* crow = ctx + (size_t)(qbase + half * 8) * DM + dbase + l15;
#pragma unroll
        for (int i = 0; i < 8; ++i) crow[(size_t)i * DM] = acc[i];
    }
}

// ---------------------------------------------------------------------------
extern "C" void kernel_launch(void* const* d_in, const int* in_sizes, int n_in,
                              void* d_out, int out_size, void* d_ws, size_t ws_size,
                              hipStream_t stream)
{
    const float*   query = (const float*)d_in[0];
    const float*   key_  = (const float*)d_in[1];
    const float*   value = (const float*)d_in[2];
    const uint8_t* mask  = (const uint8_t*)d_in[3];
    const int*     card  = (const int*)d_in[4];
    const int*     pos   = (const int*)d_in[5];
    const float*   Wq = (const float*)d_in[6];  const float* bq = (const float*)d_in[7];
    const float*   Wk = (const float*)d_in[8];  const float* bk = (const float*)d_in[9];
    const float*   Wv = (const float*)d_in[10]; const float* bv = (const float*)d_in[11];
    const float*   Wo = (const float*)d_in[12]; const float* bo = (const float*)d_in[13];

    float* out  = (float*)d_out;                         // [2048,1024]
    float* attn = out + (size_t)S_LEN * DM;              // [16,2048,2048]

    const size_t mat = (size_t)S_LEN * DM;               // 2M floats = 8MB
    float* wsQ = (float*)d_ws;
    float* wsK = wsQ + mat;
    float* wsV = wsK + mat;
    float* wsC = wsV + mat;                              // total 32MB scratch

    const int tiles  = (S_LEN / 16) * (DM / 16);         // 8192 tiles
    const int blocks = tiles / 8;                        // 1024 blocks x 8 waves

    gemm_bias_kernel<<<blocks, 256, 0, stream>>>(query, Wq, bq, wsQ, S_LEN, DM, DM);
    gemm_bias_kernel<<<blocks, 256, 0, stream>>>(key_,  Wk, bk, wsK, S_LEN, DM, DM);
    gemm_bias_kernel<<<blocks, 256, 0, stream>>>(value, Wv, bv, wsV, S_LEN, DM, DM);

    const int lds_bytes = 16 * SPAD * 4      // score/prob block (padded)
                        + S_LEN * 2 * 3      // rank/suit/pos (short)
                        + S_LEN * 4;         // mask addend   (float)
    (void)hipFuncSetAttribute(reinterpret_cast<const void*>(attn_kernel),
                              hipFuncAttributeMaxDynamicSharedMemorySize, lds_bytes);

    attn_kernel<<<dim3(S_LEN / 16, N_HEADS), 256, lds_bytes, stream>>>(
        wsQ, wsK, wsV, card, pos, mask, attn, wsC);

    gemm_bias_kernel<<<blocks, 256, 0, stream>>>(wsC, Wo, bo, out, S_LEN, DM, DM);
}